// MultiMeasure_7524782702793
// MI455X (gfx1250) — compile-verified
//
#include <hip/hip_runtime.h>
#include <math.h>

// Problem constants (from reference)
#define BATCH   32
#define HH      384          // full image H=W
#define KK      378          // crop size
#define NSH     49           // 7x7 shifts
#define NTOT    4572288.0f   // 32*1*378*378

// Tiling
#define ROWS    14           // SR rows per block (378 = 27*14)
#define TILES   27
#define PR      (ROWS + 6)   // patch rows incl. shift halo
#define PW      384          // patch cols (378 + 6)
#define THREADS 256
#define PXB     (KK * ROWS)  // 5292 pixels per block
#define FULLK   20           // 20*256 = 5120 <= 5292
#define TAILN   (PXB - FULLK * THREADS) // 172
#define NBLK    (BATCH * TILES)         // 864 blocks
#define NROW    (NSH * 5)               // 245 partial rows
#define RSTR    256                     // padded row stride (col-major part)

typedef __attribute__((ext_vector_type(2))) float v2f;
typedef __attribute__((ext_vector_type(8))) float v8f;

// ---------------------------------------------------------------------------
// Pass 1: per-block partial sums for all 49 shifts.
// part layout (column-major): part[blk * RSTR + (s*5 + q)]
// Rows NROW..RSTR-1 are zero-filled so pass 2 can load them unconditionally.
// ---------------------------------------------------------------------------
__global__ __launch_bounds__(THREADS) void mm_partial_kernel(
    const float* __restrict__ pred,
    const float* __restrict__ target,
    float* __restrict__ part) {
  __shared__ float patch[PR * PW];   // 30720 B
  __shared__ float wsum[8][5];

  const int tid = threadIdx.x;
  const int img = blockIdx.y;
  const int t0  = blockIdx.x * ROWS;                  // SR-space row offset
  const float* tgt = target + (size_t)img * HH * HH;
  const float* prd = pred   + (size_t)img * HH * HH;
  const int blk = img * TILES + blockIdx.x;

  __builtin_prefetch(prd + (size_t)(3 + t0) * HH, 0, 0);

  // Zero the pad rows of this block's partial column (rows 245..255).
  if (tid >= NROW) {  // tid in [245,256)
    part[(size_t)blk * RSTR + tid] = 0.0f;
  }

  // Stage target patch rows [t0, t0+20) x cols [0,384) into LDS (30 iters/thread)
  for (int i = tid; i < PR * PW; i += THREADS) {
    int r = i / PW;
    int c = i - r * PW;
    patch[i] = tgt[(size_t)(t0 + r) * HH + c];
  }

  // SR pixels for this tile into registers
  float sr[FULLK + 1];
  int   idx[FULLK + 1];
  const bool tail = (tid < TAILN);
#pragma unroll
  for (int k = 0; k <= FULLK; ++k) {
    if (k < FULLK || tail) {
      int p = tid + k * THREADS;        // p < PXB
      int r = p / KK;
      int c = p - r * KK;
      sr[k]  = prd[(size_t)(3 + t0 + r) * HH + (3 + c)];
      idx[k] = r * PW + c;
    } else {
      sr[k] = 0.0f;
      idx[k] = 0;
    }
  }
  __syncthreads();

  const int lane = tid & 31;
  const int wid  = tid >> 5;

  for (int s = 0; s < NSH; ++s) {
    const int u = s / 7;
    const int v = s - u * 7;
    const int off = u * PW + v;

    float c1 = 0.f, c2 = 0.f, c3 = 0.f, c4 = 0.f, c5 = 0.f;
#pragma unroll
    for (int k = 0; k <= FULLK; ++k) {
      if (k < FULLK || tail) {
        float hr = patch[idx[k] + off];
        float d  = hr - sr[k];
        float m  = (hr > 5.0f) ? 1.0f : 0.0f;
        c1 += m;
        c2 += m * d;
        c3 += d;
        c4 += d * d;
        c5 += fabsf(d);
      }
    }
    // wave32 butterfly reduction (fixed order -> deterministic)
#pragma unroll
    for (int o = 16; o > 0; o >>= 1) {
      c1 += __shfl_xor(c1, o);
      c2 += __shfl_xor(c2, o);
      c3 += __shfl_xor(c3, o);
      c4 += __shfl_xor(c4, o);
      c5 += __shfl_xor(c5, o);
    }
    if (lane == 0) {
      wsum[wid][0] = c1; wsum[wid][1] = c2; wsum[wid][2] = c3;
      wsum[wid][3] = c4; wsum[wid][4] = c5;
    }
    __syncthreads();
    if (tid == 0) {
      float t[5] = {0.f, 0.f, 0.f, 0.f, 0.f};
      for (int w = 0; w < 8; ++w)
        for (int q = 0; q < 5; ++q)
          t[q] += wsum[w][q];
      for (int q = 0; q < 5; ++q)
        part[(size_t)blk * RSTR + (s * 5 + q)] = t[q];
    }
    __syncthreads();
  }
}

// ---------------------------------------------------------------------------
// Pass 2 (single block): totals[row] = sum_blk part[blk][row] via
// V_WMMA_F32_16X16X4_F32 with B = ones (row-reduction as matrix product),
// then cmse/argmin/cPSNR/MAE.
// A layout (documented): lanes 0-15 -> M=lane, K=0(a.x)/1(a.y);
//                        lanes 16-31 -> M=lane-16, K=2(a.x)/3(a.y).
// D layout: VGPR j lanes 0-15 = row j, lanes 16-31 = row j+8 (constant over N
// since every row of D is the same row-sum replicated across columns).
// Pad rows are zero-filled by pass 1, so all loads are unconditional
// (no exec-mask branching around the loads -> clean pipelined loop).
// ---------------------------------------------------------------------------
__global__ __launch_bounds__(THREADS) void mm_finalize_kernel(
    const float* __restrict__ part,
    float* __restrict__ out) {
  __shared__ float totals[256];
  __shared__ float cms[64];

  const int tid  = threadIdx.x;
  const int lane = tid & 31;
  const int wid  = tid >> 5;

  // 16 row-groups of 16; each of the 8 waves handles 2 groups.
#pragma unroll
  for (int gi = 0; gi < 2; ++gi) {
    const int row0 = (wid + gi * 8) * 16;
    const int m    = lane & 15;
    const int kh   = (lane >> 4) << 1;      // 0 for lanes 0-15, 2 for 16-31
    const int row  = row0 + m;

    v8f acc = {0.f, 0.f, 0.f, 0.f, 0.f, 0.f, 0.f, 0.f};
    v2f ones; ones.x = 1.0f; ones.y = 1.0f;

    const float* col = part + row;          // lane-coalesced within half-wave
#pragma unroll 4
    for (int kb = 0; kb < NBLK; kb += 4) {
      v2f a;
      a.x = col[(size_t)(kb + kh)     * RSTR];
      a.y = col[(size_t)(kb + kh + 1) * RSTR];
      acc = __builtin_amdgcn_wmma_f32_16x16x4_f32(
          /*neg_a=*/false, a, /*neg_b=*/false, ones,
          /*c_mod=*/(short)0, acc, /*reuse_a=*/false, /*reuse_b=*/false);
    }
    if (lane == 0) {
#pragma unroll
      for (int j = 0; j < 8; ++j) totals[row0 + j] = acc[j];
    } else if (lane == 16) {
#pragma unroll
      for (int j = 0; j < 8; ++j) totals[row0 + 8 + j] = acc[j];
    }
  }
  __syncthreads();

  if (tid < NSH) {
    const float S1 = totals[tid * 5 + 0];
    const float S2 = totals[tid * 5 + 1];
    const float S3 = totals[tid * 5 + 2];
    const float S4 = totals[tid * 5 + 3];
    const float co = 1.0f / S1;
    const float b  = co * S2;
    cms[tid] = co * (S4 - 2.0f * b * S3 + NTOT * b * b);
  }
  __syncthreads();

  if (tid == 0) {
    int best = 0;
    float mn = cms[0];
    for (int s = 1; s < NSH; ++s) {           // strict '<' => first-index argmin
      float v = cms[s];
      if (v < mn) { mn = v; best = s; }
    }
    out[0] = -10.0f * log10f(mn);             // cPSNR
    out[1] = totals[best * 5 + 4] / NTOT;     // MAE
  }
}

// ---------------------------------------------------------------------------
extern "C" void kernel_launch(void* const* d_in, const int* in_sizes, int n_in,
                              void* d_out, int out_size, void* d_ws, size_t ws_size,
                              hipStream_t stream) {
  const float* pred   = (const float*)d_in[0];
  const float* target = (const float*)d_in[1];
  // d_in[2] (blended_target) is unused by the reference.
  float* part = (float*)d_ws;   // needs NBLK*RSTR*4 = 884,736 bytes

  dim3 grid1(TILES, BATCH);
  mm_partial_kernel<<<grid1, THREADS, 0, stream>>>(pred, target, part);
  mm_finalize_kernel<<<1, THREADS, 0, stream>>>(part, (float*)d_out);
}